// ActorCriticGNN_74534862455058
// MI455X (gfx1250) — compile-verified
//
#include <hip/hip_runtime.h>
#include <hip/hip_bf16.h>
#include <math.h>

#define N_NODES 20000
#define E_EDGES 320000
#define ETOT    (E_EDGES + N_NODES)   // self-loops appended
#define DIN     64
#define HID     128
#define NHEAD   4
#define HC      (NHEAD * HID)         // 512

typedef float v2f __attribute__((ext_vector_type(2)));
typedef float v4f __attribute__((ext_vector_type(4)));
typedef float v8f __attribute__((ext_vector_type(8)));

// ---- monotone float<->uint mapping for atomicMax-based segment max ----
__device__ __forceinline__ unsigned f2key(float f) {
    unsigned u = __float_as_uint(f);
    return (u & 0x80000000u) ? ~u : (u | 0x80000000u);
}
__device__ __forceinline__ float key2f(unsigned k) {
    unsigned u = (k & 0x80000000u) ? (k ^ 0x80000000u) : ~k;
    return __uint_as_float(u);
}

// =====================================================================
// fp32 WMMA GEMM: C[M,N] = act(A[M,K] @ B[K,N] + bias[N])
// One wave computes one 16x16 tile; K stepped by 4 via V_WMMA_F32_16X16X4_F32.
// A layout: lane%16 = M row, lane/16 selects K pair {0,1}|{2,3}, VGPR = K in pair.
// B layout: lane%16 = N col, lane/16 selects K pair, VGPR = K in pair.
// C layout: lane%16 = N col, VGPR v -> M row = v + 8*(lane/16).
// =====================================================================
template <int RELU>
__global__ void gemm_wmma_f32(const float* __restrict__ A,
                              const float* __restrict__ B,
                              const float* __restrict__ bias,
                              float* __restrict__ C,
                              int M, int K, int N) {
    int wave   = (blockIdx.x * blockDim.x + threadIdx.x) >> 5;
    int tilesN = N >> 4;
    int tiles  = (M >> 4) * tilesN;
    if (wave >= tiles) return;                 // uniform per wave; EXEC stays full
    int tm = wave / tilesN, tn = wave % tilesN;
    int lane = threadIdx.x & 31;
    int l16  = lane & 15;
    int half = lane >> 4;

    const float* arow = A + (size_t)(tm * 16 + l16) * K + 2 * half;
    const float* bcol = B + (size_t)(2 * half) * N + tn * 16 + l16;

    v8f acc = {0.f, 0.f, 0.f, 0.f, 0.f, 0.f, 0.f, 0.f};
    for (int k = 0; k < K; k += 4) {
        v2f a, b;
        a.x = arow[k];
        a.y = arow[k + 1];
        b.x = bcol[(size_t)k * N];
        b.y = bcol[(size_t)(k + 1) * N];
        acc = __builtin_amdgcn_wmma_f32_16x16x4_f32(
            /*neg_a=*/false, a, /*neg_b=*/false, b,
            /*c_mod=*/(short)0, acc, /*reuse_a=*/false, /*reuse_b=*/false);
    }

    int   col = tn * 16 + l16;
    float bv  = bias[col];
#pragma unroll
    for (int v = 0; v < 8; ++v) {
        int   row = tm * 16 + v + 8 * half;
        float val = acc[v] + bv;
        if (RELU) val = fmaxf(val, 0.f);
        C[(size_t)row * N + col] = val;
    }
}

// =====================================================================
// Per-call scratch init (harness poisons ws; must re-zero every launch)
// =====================================================================
__global__ void init_scratch(float* __restrict__ res, float* __restrict__ denom,
                             unsigned* __restrict__ amax, float* __restrict__ gpart) {
    int idx = blockIdx.x * blockDim.x + threadIdx.x;
    if (idx < N_NODES * HID) res[idx] = 0.f;
    if (idx < N_NODES * NHEAD) { denom[idx] = 0.f; amax[idx] = 0u; }  // 0 < key(-inf): <= all keys
    if (idx < HID) gpart[idx] = 0.f;
}

// ---- wave-uniform edge endpoints (SGPR-resident) ----
__device__ __forceinline__ void edge_ends(const int* __restrict__ ei, int e, int& s, int& d) {
    if (e < E_EDGES) { s = ei[e]; d = ei[E_EDGES + e]; }
    else             { s = d = e - E_EDGES; }
    s = __builtin_amdgcn_readfirstlane(s);
    d = __builtin_amdgcn_readfirstlane(d);
}

// =====================================================================
// Edge pass 1: alpha[e][h] = <leakyrelu(xl[src]+xr[dst]), att[h]>, segment max
// One wave per edge; b128 gathers (4 contiguous floats per lane per head row).
// =====================================================================
__global__ void edge_alpha(const int* __restrict__ ei,
                           const float* __restrict__ xl, const float* __restrict__ xr,
                           const float* __restrict__ att,
                           float* __restrict__ alpha, unsigned* __restrict__ amax) {
    int wave = (blockIdx.x * blockDim.x + threadIdx.x) >> 5;
    if (wave >= ETOT) return;
    int lane = threadIdx.x & 31;
    int s, d;
    edge_ends(ei, wave, s, d);

    const v4f* pl = (const v4f*)(xl + (size_t)s * HC);   // 128 v4f per row
    const v4f* pr = (const v4f*)(xr + (size_t)d * HC);
    const v4f* pa = (const v4f*)att;

    float acc[NHEAD];
#pragma unroll
    for (int h = 0; h < NHEAD; ++h) {
        int idx = h * (HID / 4) + lane;                  // lane owns 4 contiguous floats
        v4f l = pl[idx];
        v4f r = pr[idx];
        v4f w = pa[idx];
        float a = 0.f;
#pragma unroll
        for (int i = 0; i < 4; ++i) {
            float m = l[i] + r[i];
            float e = (m > 0.f) ? m : 0.2f * m;          // leaky_relu
            a += e * w[i];
        }
        acc[h] = a;
    }
#pragma unroll
    for (int h = 0; h < NHEAD; ++h)
        for (int off = 16; off; off >>= 1)
            acc[h] += __shfl_down(acc[h], off, 32);

    if (lane == 0) {
#pragma unroll
        for (int h = 0; h < NHEAD; ++h) {
            alpha[(size_t)wave * NHEAD + h] = acc[h];
            atomicMax(&amax[d * NHEAD + h], f2key(acc[h]));
        }
    }
}

// =====================================================================
// Edge pass 2: ex = exp(alpha - amax[dst]); denom[dst] += ex  (in place)
// =====================================================================
__global__ void edge_exp(const int* __restrict__ ei, float* __restrict__ alpha,
                         const unsigned* __restrict__ amax, float* __restrict__ denom) {
    int idx = blockIdx.x * blockDim.x + threadIdx.x;
    if (idx >= ETOT * NHEAD) return;
    int e = idx >> 2, h = idx & 3;
    int d = (e < E_EDGES) ? ei[E_EDGES + e] : (e - E_EDGES);
    float ex = expf(alpha[idx] - key2f(amax[d * NHEAD + h]));
    alpha[idx] = ex;
    atomicAdd(&denom[d * NHEAD + h], ex);
}

// =====================================================================
// Edge pass 3: res[dst][c] += (1/H) * sum_h xl[src][h,c] * a[e,h]
// (head-mean fused; b128 gathers, 4 atomics per lane)
// =====================================================================
__global__ void edge_scatter(const int* __restrict__ ei, const float* __restrict__ xl,
                             const float* __restrict__ alpha, const float* __restrict__ denom,
                             float* __restrict__ res) {
    int wave = (blockIdx.x * blockDim.x + threadIdx.x) >> 5;
    if (wave >= ETOT) return;
    int lane = threadIdx.x & 31;
    int s, d;
    edge_ends(ei, wave, s, d);

    float a4[NHEAD];
#pragma unroll
    for (int h = 0; h < NHEAD; ++h)
        a4[h] = alpha[(size_t)wave * NHEAD + h] / denom[d * NHEAD + h];

    const v4f* pl = (const v4f*)(xl + (size_t)s * HC);
    float* pres = res + (size_t)d * HID + lane * 4;

    v4f v = {0.f, 0.f, 0.f, 0.f};
#pragma unroll
    for (int h = 0; h < NHEAD; ++h) {
        v4f l = pl[h * (HID / 4) + lane];
#pragma unroll
        for (int i = 0; i < 4; ++i) v[i] += l[i] * a4[h];
    }
#pragma unroll
    for (int i = 0; i < 4; ++i)
        atomicAdd(&pres[i], 0.25f * v[i]);
}

// =====================================================================
// Node: hout = relu(layernorm(hin + res + conv_bias) * g + be)
// One wave per node, 4 contiguous features per lane, b128 traffic.
// =====================================================================
__global__ void node_ln_relu(const float* __restrict__ hin, const float* __restrict__ res,
                             const float* __restrict__ bias, const float* __restrict__ g,
                             const float* __restrict__ be, float* __restrict__ hout) {
    int wave = (blockIdx.x * blockDim.x + threadIdx.x) >> 5;
    if (wave >= N_NODES) return;
    int lane = threadIdx.x & 31;
    const v4f* ph = (const v4f*)(hin + (size_t)wave * HID);
    const v4f* pr = (const v4f*)(res + (size_t)wave * HID);
    const v4f* pb = (const v4f*)bias;

    v4f y;
    {
        v4f a = ph[lane], b = pr[lane], c = pb[lane];
#pragma unroll
        for (int i = 0; i < 4; ++i) y[i] = a[i] + b[i] + c[i];
    }
    float s = 0.f, ss = 0.f;
#pragma unroll
    for (int i = 0; i < 4; ++i) { s += y[i]; ss += y[i] * y[i]; }
    for (int off = 16; off; off >>= 1) {
        s  += __shfl_xor(s,  off, 32);
        ss += __shfl_xor(ss, off, 32);
    }
    float mu  = s * (1.f / HID);
    float var = ss * (1.f / HID) - mu * mu;
    float r   = rsqrtf(var + 1e-5f);

    v4f gg = ((const v4f*)g)[lane];
    v4f bb = ((const v4f*)be)[lane];
    v4f o;
#pragma unroll
    for (int i = 0; i < 4; ++i)
        o[i] = fmaxf((y[i] - mu) * r * gg[i] + bb[i], 0.f);
    ((v4f*)(hout + (size_t)wave * HID))[lane] = o;
}

// =====================================================================
// logits[n] = <shared[n], Wa> + ba ; also accumulate gpart[c] += shared[n][c]
// =====================================================================
__global__ void logits_kernel(const float* __restrict__ shared_f, const float* __restrict__ Wa,
                              const float* __restrict__ ba, float* __restrict__ gpart,
                              float* __restrict__ out) {
    int wave = (blockIdx.x * blockDim.x + threadIdx.x) >> 5;
    if (wave >= N_NODES) return;
    int lane = threadIdx.x & 31;
    v4f v = ((const v4f*)(shared_f + (size_t)wave * HID))[lane];
    v4f w = ((const v4f*)Wa)[lane];
    float acc = 0.f;
#pragma unroll
    for (int i = 0; i < 4; ++i) {
        acc += v[i] * w[i];
        atomicAdd(&gpart[lane * 4 + i], v[i]);
    }
    for (int off = 16; off; off >>= 1) acc += __shfl_down(acc, off, 32);
    if (lane == 0) out[wave] = acc + ba[0];
}

// value = <gpart/N, Wc> + bc  -> out[N_NODES]  (one wave)
__global__ void value_kernel(const float* __restrict__ gpart, const float* __restrict__ Wc,
                             const float* __restrict__ bc, float* __restrict__ out) {
    int lane = threadIdx.x & 31;
    v4f v = ((const v4f*)gpart)[lane];
    v4f w = ((const v4f*)Wc)[lane];
    float acc = 0.f;
#pragma unroll
    for (int i = 0; i < 4; ++i) acc += (v[i] * (1.f / N_NODES)) * w[i];
    for (int off = 16; off; off >>= 1) acc += __shfl_down(acc, off, 32);
    if (lane == 0) out[N_NODES] = acc + bc[0];
}

// =====================================================================
static inline int blocksForWaves(long long nwaves) {
    long long threads = nwaves * 32;
    return (int)((threads + 255) / 256);
}

extern "C" void kernel_launch(void* const* d_in, const int* in_sizes, int n_in,
                              void* d_out, int out_size, void* d_ws, size_t ws_size,
                              hipStream_t stream) {
    const float* x     = (const float*)d_in[0];
    const int*   ei    = (const int*)d_in[1];
    const float* W_in  = (const float*)d_in[2];
    const float* b_in  = (const float*)d_in[3];
    const float* Wl1   = (const float*)d_in[4];
    const float* bl1   = (const float*)d_in[5];
    const float* Wr1   = (const float*)d_in[6];
    const float* br1   = (const float*)d_in[7];
    const float* att1  = (const float*)d_in[8];
    const float* bias1 = (const float*)d_in[9];
    const float* g1    = (const float*)d_in[10];
    const float* be1   = (const float*)d_in[11];
    const float* Wl2   = (const float*)d_in[12];
    const float* bl2   = (const float*)d_in[13];
    const float* Wr2   = (const float*)d_in[14];
    const float* br2   = (const float*)d_in[15];
    const float* att2  = (const float*)d_in[16];
    const float* bias2 = (const float*)d_in[17];
    const float* g2    = (const float*)d_in[18];
    const float* be2   = (const float*)d_in[19];
    const float* Wa    = (const float*)d_in[20];
    const float* ba    = (const float*)d_in[21];
    const float* Wc    = (const float*)d_in[22];
    const float* bc    = (const float*)d_in[23];

    float* out = (float*)d_out;   // [0..N-1] logits, [N] value

    // ---- carve workspace ----
    float* ws = (float*)d_ws;
    size_t off = 0;
    float*    h0    = ws + off; off += (size_t)N_NODES * HID;   // input-proj / shared
    float*    h1    = ws + off; off += (size_t)N_NODES * HID;   // after block1
    float*    xl    = ws + off; off += (size_t)N_NODES * HC;
    float*    xr    = ws + off; off += (size_t)N_NODES * HC;
    float*    alpha = ws + off; off += (size_t)ETOT * NHEAD;    // alpha -> ex (in place)
    unsigned* amax  = (unsigned*)(ws + off); off += (size_t)N_NODES * NHEAD;
    float*    denom = ws + off; off += (size_t)N_NODES * NHEAD;
    float*    res   = ws + off; off += (size_t)N_NODES * HID;
    float*    gpart = ws + off; off += HID;
    (void)ws_size; (void)in_sizes; (void)n_in; (void)out_size;

    const int initBlocks  = (N_NODES * HID + 255) / 256;
    const int edgeBlocks  = blocksForWaves(ETOT);
    const int ehBlocks    = (ETOT * NHEAD + 255) / 256;
    const int nodeBlocks  = blocksForWaves(N_NODES);
    const int gemmInBlk   = blocksForWaves((N_NODES / 16) * (HID / 16));   // 10000 waves
    const int gemmHCBlk   = blocksForWaves((N_NODES / 16) * (HC / 16));    // 40000 waves

    // ---- input projection: h0 = relu(x @ W_in + b_in) ----
    gemm_wmma_f32<1><<<gemmInBlk, 256, 0, stream>>>(x, W_in, b_in, h0, N_NODES, DIN, HID);

    // ================= conv1 =================
    init_scratch<<<initBlocks, 256, 0, stream>>>(res, denom, amax, gpart);
    gemm_wmma_f32<0><<<gemmHCBlk, 256, 0, stream>>>(h0, Wl1, bl1, xl, N_NODES, HID, HC);
    gemm_wmma_f32<0><<<gemmHCBlk, 256, 0, stream>>>(h0, Wr1, br1, xr, N_NODES, HID, HC);
    edge_alpha  <<<edgeBlocks, 256, 0, stream>>>(ei, xl, xr, att1, alpha, amax);
    edge_exp    <<<ehBlocks,   256, 0, stream>>>(ei, alpha, amax, denom);
    edge_scatter<<<edgeBlocks, 256, 0, stream>>>(ei, xl, alpha, denom, res);
    node_ln_relu<<<nodeBlocks, 256, 0, stream>>>(h0, res, bias1, g1, be1, h1);

    // ================= conv2 =================
    init_scratch<<<initBlocks, 256, 0, stream>>>(res, denom, amax, gpart);
    gemm_wmma_f32<0><<<gemmHCBlk, 256, 0, stream>>>(h1, Wl2, bl2, xl, N_NODES, HID, HC);
    gemm_wmma_f32<0><<<gemmHCBlk, 256, 0, stream>>>(h1, Wr2, br2, xr, N_NODES, HID, HC);
    edge_alpha  <<<edgeBlocks, 256, 0, stream>>>(ei, xl, xr, att2, alpha, amax);
    edge_exp    <<<ehBlocks,   256, 0, stream>>>(ei, alpha, amax, denom);
    edge_scatter<<<edgeBlocks, 256, 0, stream>>>(ei, xl, alpha, denom, res);
    node_ln_relu<<<nodeBlocks, 256, 0, stream>>>(h1, res, bias2, g2, be2, h0);  // h0 = shared

    // ================= heads =================
    logits_kernel<<<nodeBlocks, 256, 0, stream>>>(h0, Wa, ba, gpart, out);
    value_kernel <<<1, 32, 0, stream>>>(gpart, Wc, bc, out);
}